// TreeConv1d_67619965108945
// MI455X (gfx1250) — compile-verified
//
#include <hip/hip_runtime.h>

typedef float v2f __attribute__((ext_vector_type(2)));
typedef float v8f __attribute__((ext_vector_type(8)));

#define Bsz  64
#define Cch  256
#define Och  256
#define Npt  512
#define Mi   1533        // 3*(N-1)
#define Kout 511         // Mi/3 real output columns (col 0 of out is zeros)
#define CT   768         // C*3 contraction depth
#define KT   16          // k-tile width
#define NKT  32          // ceil(Kout/KT)

// One block = one (batch, k-tile). 8 waves, each wave owns two 16-row o-tiles.
// LDS layout: G rows pair-interleaved so a lane's B fragment is one b64:
//   sG[(r>>1)*32 + k*2 + (r&1)] = G[r][k]
__global__ __launch_bounds__(256)
void treeconv_wmma_f32(const float* __restrict__ data,
                       const long long* __restrict__ indexes,
                       const float* __restrict__ W,
                       const float* __restrict__ bias,
                       float* __restrict__ out) {
    __shared__ float sG[CT * KT];   // 48 KB
    __shared__ int   sIdx[3 * KT];

    const int tid   = threadIdx.x;
    const int b     = blockIdx.x / NKT;
    const int ktile = blockIdx.x % NKT;

    // ---- stage the 48 gather indexes for this k-tile ----
    if (tid < 3 * KT) {
        int gi = ktile * (3 * KT) + tid;            // index into [Mi]
        sIdx[tid] = (gi < Mi) ? (int)indexes[(size_t)b * Mi + gi] : 0;
    }
    __syncthreads();

    // ---- cooperative gather into pair-interleaved LDS layout ----
    // element i: p=i&1, k=(i>>1)&15, j=i>>5  ->  G[2j+p][k]
    const float* dbase = data + (size_t)b * Cch * Npt;
    for (int i = tid; i < CT * KT; i += 256) {
        int p  = i & 1;
        int k  = (i >> 1) & (KT - 1);
        int ct = ((i >> 5) << 1) + p;
        int c  = ct / 3;
        int t  = ct - 3 * c;
        int kg = ktile * KT + k;
        float v = 0.0f;
        if (kg < Kout) v = dbase[c * Npt + sIdx[3 * k + t]];
        sG[i] = v;                                   // linear store, conflict-free
    }
    __syncthreads();

    // ---- WMMA loop: D(16x16) += A(16x4) x B(4x16), 192 steps over ct ----
    const int wave   = tid >> 5;                     // 0..7
    const int lane   = tid & 31;
    const int lmod   = lane & 15;
    const int khalf  = (lane >> 4) << 1;             // 0 or 2 (K sub-pair)
    const int o_base = wave * 32;                    // two o-tiles: o_base, o_base+16

    const float* aptr0 = W + (size_t)(o_base + lmod) * CT + khalf;  // A row, tile 0
    const float* aptr1 = aptr0 + 16 * CT;                            // A row, tile 1
    const int    gbase = khalf * KT + lmod * 2;      // lane's b64 slot in a row-pair

    v8f acc0 = {};
    v8f acc1 = {};
#pragma unroll 4
    for (int ct = 0; ct < CT; ct += 4) {
        v2f a0 = *(const v2f*)(aptr0 + ct);          // global_load_b64 (L2 hit)
        v2f a1 = *(const v2f*)(aptr1 + ct);
        v2f bf = *(const v2f*)(&sG[ct * KT + gbase]);  // one ds_load_b64
        acc0 = __builtin_amdgcn_wmma_f32_16x16x4_f32(
                   false, a0, false, bf, (short)0, acc0, false, false);
        acc1 = __builtin_amdgcn_wmma_f32_16x16x4_f32(
                   false, a1, false, bf, (short)0, acc1, false, false);
    }

    // ---- epilogue: bias add + store (D: VGPR r -> row r / r+8, lane -> col) ----
    const int kcol  = lmod;
    const int kg    = ktile * KT + kcol;
    const int ohalf = (lane >> 4) * 8;
    if (kg < Kout) {
        const int col = kg + 1;                      // column 0 is the zero pad
#pragma unroll
        for (int r = 0; r < 8; ++r) {
            int o0 = o_base + r + ohalf;
            out[((size_t)b * Och + o0) * Npt + col] = acc0[r] + bias[o0];
            int o1 = o0 + 16;
            out[((size_t)b * Och + o1) * Npt + col] = acc1[r] + bias[o1];
        }
    }
    // zero column kk = 0 (one thread per output row)
    if (ktile == 0) {
        out[((size_t)b * Och + tid) * Npt] = 0.0f;
    }
}

extern "C" void kernel_launch(void* const* d_in, const int* in_sizes, int n_in,
                              void* d_out, int out_size, void* d_ws, size_t ws_size,
                              hipStream_t stream) {
    (void)in_sizes; (void)n_in; (void)d_ws; (void)ws_size;

    const float*     data    = (const float*)d_in[0];
    const long long* indexes = (const long long*)d_in[1];   // int64 in reference
    const float*     W       = (const float*)d_in[2];
    const float*     bias    = (const float*)d_in[3];
    float*           out     = (float*)d_out;

    dim3 grid(Bsz * NKT);       // 2048 blocks
    dim3 block(256);            // 8 wave32 per block
    treeconv_wmma_f32<<<grid, block, 0, stream>>>(data, indexes, W, bias, out);

    // Tuple output: append the pass-through int64 indexes after the f32 feats.
    const size_t featsElems = (size_t)Bsz * Och * Npt;        // 8,388,608 f32
    const size_t idxBytes   = (size_t)Bsz * Mi * sizeof(long long);
    if ((size_t)out_size * sizeof(float) >= featsElems * sizeof(float) + idxBytes) {
        hipMemcpyAsync((char*)d_out + featsElems * sizeof(float),
                       (const void*)indexes, idxBytes,
                       hipMemcpyDeviceToDevice, stream);
    }
}